// GlobalKGHeteroGAT_10840497455104
// MI455X (gfx1250) — compile-verified
//
#include <hip/hip_runtime.h>
#include <hip/hip_bf16.h>
#include <math.h>

#define NE_N 50000
#define NM_N 50000
#define E_N  500000
#define HID  128
#define HEADS 4
#define DPH  32

typedef __attribute__((ext_vector_type(2))) float v2f;
typedef __attribute__((ext_vector_type(8))) float v8f;

// ---- monotonic float<->uint encoding for atomic max on floats ----
__device__ __forceinline__ unsigned f2ord(float f) {
    unsigned u = __float_as_uint(f);
    return (u & 0x80000000u) ? ~u : (u | 0x80000000u);
}
__device__ __forceinline__ float ord2f(unsigned u) {
    return (u & 0x80000000u) ? __uint_as_float(u & 0x7fffffffu)
                             : __uint_as_float(~u);
}
#define ORD_NEG_INF 0x007FFFFFu   // f2ord(-inf)

// =====================================================================
// WMMA fp32 GEMM: C[M,N] = act(A[M,128] @ B[128,N] + bias)
// blockDim = 128 (4 waves); wave w computes tile (blockIdx.x, blockIdx.y*4+w).
// A tile (16x128) and B panel (128x64) staged in LDS with bank-conflict
// padding; N is a compile-time constant so all inner-loop operand reads are
// ds_loads with immediate offsets (no per-iteration address math).
// ACT: 0 = none, 1 = relu
// =====================================================================
template <int N, int ACT>
__global__ void __launch_bounds__(128)
gemm128_wmma(const float* __restrict__ A, const float* __restrict__ B,
             const float* __restrict__ bias, float* __restrict__ C) {
    constexpr int AST = 132;   // A row stride (floats): 128 + 4 pad
    constexpr int BST = 72;    // B row stride (floats): 64 + 8 pad
    __shared__ float As[16 * AST];
    __shared__ float Bs[128 * BST];

    const int tid  = threadIdx.x;
    const int lane = tid & 31;
    const int wave = tid >> 5;
    const int bm   = blockIdx.x;
    const int col0 = blockIdx.y * 64;

    // cooperative, coalesced load of the 16x128 A tile
    const float* Ab = A + (size_t)bm * 16 * 128;
#pragma unroll 4
    for (int i = 0; i < 16; ++i) {
        int idx = tid + i * 128;            // 0..2047
        As[(idx >> 7) * AST + (idx & 127)] = Ab[idx];
    }
    // cooperative load of the 128x64 B panel (weights; L2-resident)
    const float* Bb = B + col0;
#pragma unroll 4
    for (int i = 0; i < 64; ++i) {
        int idx = tid + i * 128;            // 0..8191
        int k = idx >> 6, c = idx & 63;
        Bs[k * BST + c] = Bb[(size_t)k * N + c];
    }
    __syncthreads();

    const int cl    = lane & 15;
    const int colL  = wave * 16 + cl;               // local column 0..63
    const int khalf = (lane < 16) ? 0 : 2;          // ISA A/B fragment layout
    const float* ap = &As[cl * AST + khalf];
    const float* bp = &Bs[khalf * BST + colL];

    v8f acc;
    const float binit = bias ? bias[col0 + colL] : 0.0f;
#pragma unroll
    for (int r = 0; r < 8; ++r) acc[r] = binit;

#pragma unroll
    for (int k0 = 0; k0 < 128; k0 += 4) {
        v2f a, b;
        a[0] = ap[k0];
        a[1] = ap[k0 + 1];
        b[0] = bp[k0 * BST];
        b[1] = bp[k0 * BST + BST];
        acc = __builtin_amdgcn_wmma_f32_16x16x4_f32(
            false, a, false, b, (short)0, acc, false, false);
    }

    const int rbase = bm * 16 + ((lane < 16) ? 0 : 8);
    const int col   = col0 + colL;
#pragma unroll
    for (int r = 0; r < 8; ++r) {
        float v = acc[r];
        if (ACT == 1) v = fmaxf(v, 0.0f);
        C[(size_t)(rbase + r) * N + col] = v;
    }
}

// al[n,h] = sum_d hs[n, h*32+d] * a[h*32+d]
__global__ void attn_logits_kernel(const float* __restrict__ hs,
                                   const float* __restrict__ a,
                                   float* __restrict__ al, int total) {
    int idx = blockIdx.x * blockDim.x + threadIdx.x;
    if (idx >= total) return;
    int n = idx >> 2, h = idx & 3;
    const float* r  = hs + (size_t)n * HID + h * DPH;
    const float* av = a + h * DPH;
    float s = 0.0f;
#pragma unroll
    for (int d = 0; d < DPH; ++d) s += r[d] * av[d];
    al[idx] = s;
}

// zero agg/den, set m = -inf (encoded)
__global__ void conv_init_kernel(unsigned* __restrict__ mEnc,
                                 float* __restrict__ den,
                                 float* __restrict__ agg, int Nd) {
    int idx = blockIdx.x * blockDim.x + threadIdx.x;
    if (idx < Nd * HID) agg[idx] = 0.0f;
    if (idx < Nd * HEADS) { den[idx] = 0.0f; mEnc[idx] = ORD_NEG_INF; }
}

// pass 1: per (edge, head) -> atomic max of leaky-relu logit into m[dst,h]
__global__ void edge_max_kernel(const float* __restrict__ al_s,
                                const float* __restrict__ al_d,
                                unsigned* __restrict__ mEnc,
                                const int* __restrict__ src,
                                const int* __restrict__ dst, int nE) {
    int idx = blockIdx.x * blockDim.x + threadIdx.x;
    if (idx >= nE * HEADS) return;
    int e = idx >> 2, h = idx & 3;
    int s = src[e], d = dst[e];
    float v = al_s[s * HEADS + h] + al_d[d * HEADS + h];
    v = (v > 0.0f) ? v : 0.2f * v;
    atomicMax(&mEnc[d * HEADS + h], f2ord(v));
}

// pass 2: one wave32 per edge. lanes 0..3 compute exp(e-m) per head and
// accumulate den; all 32 lanes scatter the 128-float message with native
// fp32 atomics (global_atomic_add_f32).
__global__ void __launch_bounds__(256)
edge_acc_kernel(const float* __restrict__ hs,
                const float* __restrict__ al_s,
                const float* __restrict__ al_d,
                const unsigned* __restrict__ mEnc,
                float* __restrict__ den, float* __restrict__ agg,
                const int* __restrict__ src, const int* __restrict__ dst,
                int nE) {
    const int lane = threadIdx.x & 31;
    const int e = blockIdx.x * (blockDim.x >> 5) + (threadIdx.x >> 5);
    if (e >= nE) return;                    // wave-uniform exit
    const int s = src[e], d = dst[e];

    float ex = 0.0f;
    if (lane < HEADS) {
        float v = al_s[s * HEADS + lane] + al_d[d * HEADS + lane];
        v = (v > 0.0f) ? v : 0.2f * v;
        ex = expf(v - ord2f(mEnc[d * HEADS + lane]));
        unsafeAtomicAdd(&den[d * HEADS + lane], ex);
    }
    const float ex0 = __shfl(ex, 0, 32);
    const float ex1 = __shfl(ex, 1, 32);
    const float ex2 = __shfl(ex, 2, 32);
    const float ex3 = __shfl(ex, 3, 32);

    const float* hrow = hs + (size_t)s * HID;
    float* arow = agg + (size_t)d * HID;
    unsafeAtomicAdd(&arow[lane],      ex0 * hrow[lane]);
    unsafeAtomicAdd(&arow[32 + lane], ex1 * hrow[32 + lane]);
    unsafeAtomicAdd(&arow[64 + lane], ex2 * hrow[64 + lane]);
    unsafeAtomicAdd(&arow[96 + lane], ex3 * hrow[96 + lane]);
}

// epilogue: out = elu(agg/(den+1e-16) + bias)
__global__ void conv_post_kernel(const float* __restrict__ agg,
                                 const float* __restrict__ den,
                                 const float* __restrict__ bias,
                                 float* __restrict__ out, int Nd) {
    int idx = blockIdx.x * blockDim.x + threadIdx.x;
    if (idx >= Nd * HID) return;
    int n = idx >> 7, c = idx & 127, h = c >> 5;
    float v = agg[idx] / (den[n * HEADS + h] + 1e-16f) + bias[c];
    out[idx] = (v > 0.0f) ? v : expm1f(v);
}

// pred = h2[NE,64] @ Wr3[64] + br3
__global__ void pred_kernel(const float* __restrict__ h2,
                            const float* __restrict__ W,
                            const float* __restrict__ b,
                            float* __restrict__ out, int M) {
    int n = blockIdx.x * blockDim.x + threadIdx.x;
    if (n >= M) return;
    const float* r = h2 + (size_t)n * 64;
    float s = b[0];
#pragma unroll
    for (int i = 0; i < 64; ++i) s += r[i] * W[i];
    out[n] = s;
}

extern "C" void kernel_launch(void* const* d_in, const int* in_sizes, int n_in,
                              void* d_out, int out_size, void* d_ws, size_t ws_size,
                              hipStream_t stream) {
    const float* x_exp   = (const float*)d_in[0];
    const float* x_mat   = (const float*)d_in[1];
    const int*   ei_em   = (const int*)d_in[2];
    const int*   ei_me   = (const int*)d_in[3];
    const float* Win_exp = (const float*)d_in[4];
    const float* bin_exp = (const float*)d_in[5];
    const float* Win_mat = (const float*)d_in[6];
    const float* bin_mat = (const float*)d_in[7];
    const float* Wsrc_em = (const float*)d_in[8];
    const float* Wdst_em = (const float*)d_in[9];
    const float* asrc_em = (const float*)d_in[10];
    const float* adst_em = (const float*)d_in[11];
    const float* b_em    = (const float*)d_in[12];
    const float* Wsrc_me = (const float*)d_in[13];
    const float* Wdst_me = (const float*)d_in[14];
    const float* asrc_me = (const float*)d_in[15];
    const float* adst_me = (const float*)d_in[16];
    const float* b_me    = (const float*)d_in[17];
    const float* Wr1 = (const float*)d_in[18];
    const float* br1 = (const float*)d_in[19];
    const float* Wr2 = (const float*)d_in[20];
    const float* br2 = (const float*)d_in[21];
    const float* Wr3 = (const float*)d_in[22];
    const float* br3 = (const float*)d_in[23];

    const int* se = ei_em;          // exp -> mat
    const int* de = ei_em + E_N;
    const int* sm = ei_me;          // mat -> exp
    const int* dm = ei_me + E_N;

    // workspace layout (floats)
    float* ws  = (float*)d_ws;
    float* he  = ws;                      // NE x 128
    float* hm  = he  + (size_t)NE_N * HID;
    float* he2 = hm  + (size_t)NM_N * HID;
    float* hm2 = he2 + (size_t)NE_N * HID;
    float* hsb = hm2 + (size_t)NM_N * HID;   // 50000 x 128 scratch (src proj / h1)
    float* hdb = hsb + (size_t)50000 * HID;  // 50000 x 128 scratch (dst proj / h2)
    float* agg = hdb + (size_t)50000 * HID;  // 50000 x 128
    float* al_s = agg + (size_t)50000 * HID; // 50000 x 4
    float* al_d = al_s + 50000 * HEADS;
    float* den  = al_d + 50000 * HEADS;
    unsigned* mEnc = (unsigned*)(den + 50000 * HEADS);

    auto gemm = [&](const float* A, const float* B, const float* bias,
                    float* C, int M, int N, int act) {
        dim3 grid(M / 16, N / 64);
        if (N == 128) {
            if (act) gemm128_wmma<128, 1><<<grid, 128, 0, stream>>>(A, B, bias, C);
            else     gemm128_wmma<128, 0><<<grid, 128, 0, stream>>>(A, B, bias, C);
        } else {
            if (act) gemm128_wmma<64, 1><<<grid, 128, 0, stream>>>(A, B, bias, C);
            else     gemm128_wmma<64, 0><<<grid, 128, 0, stream>>>(A, B, bias, C);
        }
    };

    auto run_conv = [&](const float* h_src, const float* h_dst, int Ns, int Nd,
                        const float* Wsrc, const float* Wdst,
                        const float* asrc, const float* adst,
                        const float* bias, const int* src, const int* dst,
                        float* out_h) {
        gemm(h_src, Wsrc, nullptr, hsb, Ns, HID, 0);
        gemm(h_dst, Wdst, nullptr, hdb, Nd, HID, 0);
        attn_logits_kernel<<<(Ns * HEADS + 255) / 256, 256, 0, stream>>>(hsb, asrc, al_s, Ns * HEADS);
        attn_logits_kernel<<<(Nd * HEADS + 255) / 256, 256, 0, stream>>>(hdb, adst, al_d, Nd * HEADS);
        conv_init_kernel<<<(Nd * HID + 255) / 256, 256, 0, stream>>>(mEnc, den, agg, Nd);
        edge_max_kernel<<<(E_N * HEADS + 255) / 256, 256, 0, stream>>>(al_s, al_d, mEnc, src, dst, E_N);
        edge_acc_kernel<<<(E_N + 7) / 8, 256, 0, stream>>>(hsb, al_s, al_d, mEnc, den, agg, src, dst, E_N);
        conv_post_kernel<<<(Nd * HID + 255) / 256, 256, 0, stream>>>(agg, den, bias, out_h, Nd);
    };

    // input projections
    gemm(x_exp, Win_exp, bin_exp, he, NE_N, HID, 0);
    gemm(x_mat, Win_mat, bin_mat, hm, NM_N, HID, 0);

    // layer 1 (both convs read old he/hm; write double buffers)
    run_conv(he, hm, NE_N, NM_N, Wsrc_em, Wdst_em, asrc_em, adst_em, b_em,
             se, de, hm2);
    run_conv(hm, he, NM_N, NE_N, Wsrc_me, Wdst_me, asrc_me, adst_me, b_me,
             sm, dm, he2);

    // layer 2 writes final he/hm directly into d_out
    float* pred   = (float*)d_out;
    float* he_out = pred + NE_N;
    float* hm_out = he_out + (size_t)NE_N * HID;
    const size_t WOFF = (size_t)HID * HID;   // layer stride for W (128*128)
    const size_t AOFF = (size_t)HEADS * DPH; // layer stride for a (128)
    run_conv(he2, hm2, NE_N, NM_N, Wsrc_em + WOFF, Wdst_em + WOFF,
             asrc_em + AOFF, adst_em + AOFF, b_em + HID, se, de, hm_out);
    run_conv(hm2, he2, NM_N, NE_N, Wsrc_me + WOFF, Wdst_me + WOFF,
             asrc_me + AOFF, adst_me + AOFF, b_me + HID, sm, dm, he_out);

    // regressor MLP on experiment nodes
    gemm(he_out, Wr1, br1, hsb, NE_N, HID, 1);   // relu
    gemm(hsb,    Wr2, br2, hdb, NE_N, 64, 1);    // relu
    pred_kernel<<<(NE_N + 255) / 256, 256, 0, stream>>>(hdb, Wr3, br3, pred, NE_N);
}